// IF_88957362634870
// MI455X (gfx1250) — compile-verified
//
#include <hip/hip_runtime.h>

// Integrate-and-fire spike re-encoder (forward only).
// Purely elementwise over the frame (B,C,H,W); recurrent over time in registers.
// HBM-bound: 128 MiB in + 128 MiB out => ~11 us floor at 23.3 TB/s on MI455X.

typedef float v4f __attribute__((ext_vector_type(4)));

constexpr int   kT1  = 8;
constexpr int   kT2  = 8;
constexpr int   kT3  = 8;          // max(T1, T2)
constexpr float kEPS = -8e-05f;

__global__ __launch_bounds__(256) void if_spike_kernel(
    const float* __restrict__ x,        // (T1, F) flat, F = frame elements
    const float* __restrict__ thresh,   // scalar param
    float* __restrict__ out,            // (T2, F) flat
    long long n4)                       // F / 4 (float4 chunks per frame)
{
    long long i4 = (long long)blockIdx.x * blockDim.x + threadIdx.x;
    if (i4 >= n4) return;

    const float thre = thresh[0] * 0.125f;   // thresh / L, L = 8

    const v4f* __restrict__ xv = (const v4f*)x;
    v4f* __restrict__ ov = (v4f*)out;

    // Stage all 8 time-steps: 8 independent b128 NT loads in flight per lane.
    v4f xt[kT1];
#pragma unroll
    for (int t = 0; t < kT1; ++t)
        xt[t] = __builtin_nontemporal_load(&xv[(long long)t * n4 + i4]);

    float mem[4], sc[4];
#pragma unroll
    for (int k = 0; k < 4; ++k) { mem[k] = 0.5f * thre; sc[k] = 0.0f; }

    // Phase 1: integrate-and-fire over T1 input frames.
#pragma unroll
    for (int t = 0; t < kT1; ++t) {
#pragma unroll
        for (int k = 0; k < 4; ++k) {
            float m     = mem[k] + xt[t][k];
            float spike = ((m - thre) >= kEPS) ? thre : 0.0f;
            mem[k] = m - spike;
            sc[k]  = sc[k] + spike;
        }
    }

    // Phase 2: T3-1 relaxation steps with reverse spikes.
#pragma unroll
    for (int s = 0; s < kT3 - 1; ++s) {
#pragma unroll
        for (int k = 0; k < 4; ++k) {
            float m     = mem[k];
            float spike = ((m - thre) >= kEPS) ? thre : 0.0f;
            float rev   = (-m > 0.0f) ? thre : 0.0f;
            mem[k] = m - spike + rev;
            sc[k]  = (sc[k] + spike) - rev;
        }
    }

    // Phase 3: re-emit spike count as T2 output frames.
#pragma unroll
    for (int t = 0; t < kT2; ++t) {
        v4f spikev;
#pragma unroll
        for (int k = 0; k < 4; ++k) {
            float m     = sc[k];
            float spike = ((m - thre) >= kEPS) ? thre : 0.0f;
            spikev[k] = spike;
            sc[k]     = m - spike;
        }
        __builtin_nontemporal_store(spikev, &ov[(long long)t * n4 + i4]);
    }
}

extern "C" void kernel_launch(void* const* d_in, const int* in_sizes, int n_in,
                              void* d_out, int out_size, void* d_ws, size_t ws_size,
                              hipStream_t stream) {
    const float* x      = (const float*)d_in[0];
    const float* thresh = (const float*)d_in[1];
    float*       out    = (float*)d_out;

    const long long total = (long long)in_sizes[0];   // T1 * F
    const long long F     = total / kT1;              // frame element count
    const long long n4    = F / 4;                    // float4 chunks (F % 4 == 0)

    const int  block = 256;                           // 8 wave32s per block
    const long long grid = (n4 + block - 1) / block;  // 4096 blocks for ref shape

    if_spike_kernel<<<(unsigned)grid, block, 0, stream>>>(x, thresh, out, n4);
}